// get_k_nearest_neighbours_54503134986774
// MI455X (gfx1250) — compile-verified
//
#include <hip/hip_runtime.h>
#include <math.h>

typedef __attribute__((ext_vector_type(2))) float v2f;
typedef __attribute__((ext_vector_type(8))) float v8f;

#define N_ROWS 100000
#define DIM    2048
#define KSEL   50
#define RSLICES 128
#define ROWS_PER_SLICE 782   // ceil(100000/128)
#define SLICE  2000
#define NBLK_A 50            // 50 * 2000 = 100000 exactly
#define NCAND  (NBLK_A * KSEL)

// ---------------------------------------------------------------------------
// Kernel 1: partial column sums of squares. Block = 256 threads covering 256
// consecutive columns (coalesced 1KB per row); grid.y slices the rows.
// ---------------------------------------------------------------------------
__global__ void __launch_bounds__(256)
colsumsq_partial(const float* __restrict__ x, float* __restrict__ partial) {
    const int col = blockIdx.x * 256 + threadIdx.x;
    const int r0  = blockIdx.y * ROWS_PER_SLICE;
    const int r1  = min(r0 + ROWS_PER_SLICE, N_ROWS);
    float acc = 0.0f;
    for (int r = r0; r < r1; ++r) {
        float val = __builtin_nontemporal_load(&x[(size_t)r * DIM + col]);
        acc = fmaf(val, val, acc);
    }
    partial[(size_t)blockIdx.y * DIM + col] = acc;
}

// ---------------------------------------------------------------------------
// Kernel 2: reduce partials -> w_d = t_d / (||t|| * c_d), v_d = 1 / c_d^2.
// Single block, fixed-order reductions (deterministic).
// ---------------------------------------------------------------------------
__global__ void __launch_bounds__(256)
finalize_wv(const float* __restrict__ partial, const float* __restrict__ test,
            float* __restrict__ w, float* __restrict__ v) {
    __shared__ float red[256];
    const int t = threadIdx.x;
    float s = 0.0f;
    for (int c = t; c < DIM; c += 256) { float q = test[c]; s = fmaf(q, q, s); }
    red[t] = s;
    __syncthreads();
    for (int off = 128; off > 0; off >>= 1) {
        if (t < off) red[t] += red[t + off];
        __syncthreads();
    }
    const float inv_tnorm = rsqrtf(red[0]);
    for (int c = t; c < DIM; c += 256) {
        float cs = 0.0f;
        for (int ry = 0; ry < RSLICES; ++ry) cs += partial[(size_t)ry * DIM + c];
        float invc = rsqrtf(cs);
        w[c] = test[c] * invc * inv_tnorm;
        v[c] = invc * invc;
    }
}

// ---------------------------------------------------------------------------
// Kernel 3: distances via V_WMMA_F32_16X16X4_F32.
// dist_i^2 = 1 - 2*(x_i . w) + (x_i^2 . v).
// A 16x4 fp32 tile: lane L holds row (L&15), K-pair 2*(L>>4) -> contiguous
// float2 load. B 4x16: same float2 of w (or v) broadcast across all 16
// columns, so every D-VGPR holds the row dot replicated.
// ---------------------------------------------------------------------------
__global__ void __launch_bounds__(256)
dist_wmma(const float* __restrict__ x, const float* __restrict__ wv_w,
          const float* __restrict__ wv_v, float* __restrict__ dist) {
    __shared__ float wl[DIM];
    __shared__ float vl[DIM];
    const int t = threadIdx.x;
    for (int c = t; c < DIM; c += 256) { wl[c] = wv_w[c]; vl[c] = wv_v[c]; }
    __syncthreads();

    const int lane    = t & 31;
    const int wave    = t >> 5;
    const int half    = lane >> 4;               // 0: K{0,1}, 1: K{2,3}
    const int rowtile = blockIdx.x * 128 + wave * 16;
    const int rowc    = min(rowtile + (lane & 15), N_ROWS - 1); // clamp: keep EXEC all-1s
    const float* rp   = x + (size_t)rowc * DIM + 2 * half;

    v8f acc1 = {};   // x . w   (broadcast across N)
    v8f acc2 = {};   // x^2 . v

#pragma unroll 4
    for (int j = 0; j < DIM / 4; ++j) {
        v2f a  = __builtin_nontemporal_load((const v2f*)(rp + j * 4));
        v2f bw = *(const v2f*)&wl[j * 4 + 2 * half];
        v2f bv = *(const v2f*)&vl[j * 4 + 2 * half];
        v2f a2 = a * a;
        acc1 = __builtin_amdgcn_wmma_f32_16x16x4_f32(false, a,  false, bw,
                                                     (short)0, acc1, false, false);
        acc2 = __builtin_amdgcn_wmma_f32_16x16x4_f32(false, a2, false, bv,
                                                     (short)0, acc2, false, false);
    }

    // D layout: VGPR r, lanes 0-15 -> M=r ; lanes 16-31 -> M=8+r. Values are
    // replicated across N, so lanes 0 and 16 hold everything we need.
    if ((lane & 15) == 0) {
#pragma unroll
        for (int r = 0; r < 8; ++r) {
            int row = rowtile + half * 8 + r;
            if (row < N_ROWS) {
                float d2 = 1.0f - 2.0f * acc1[r] + acc2[r];
                dist[row] = sqrtf(fmaxf(d2, 0.0f));
            }
        }
    }
}

// ---------------------------------------------------------------------------
// Kernel 4a: per-slice top-50 by iterative argmin over an LDS-resident slice.
// Stable tie-break on smaller index (matches jax.lax.top_k stability).
// ---------------------------------------------------------------------------
__global__ void __launch_bounds__(256)
topk_phase_a(const float* __restrict__ dist, float* __restrict__ candVal,
             int* __restrict__ candIdx) {
    __shared__ float vals[SLICE];
    __shared__ float rv[256];
    __shared__ int   ri[256];
    const int t    = threadIdx.x;
    const int base = blockIdx.x * SLICE;
    for (int i = t; i < SLICE; i += 256) vals[i] = dist[base + i];
    __syncthreads();

    for (int iter = 0; iter < KSEL; ++iter) {
        float bv = __builtin_inff();
        int   bi = 0x7fffffff;
        for (int i = t; i < SLICE; i += 256) {
            float vq = vals[i];
            if (vq < bv || (vq == bv && i < bi)) { bv = vq; bi = i; }
        }
        rv[t] = bv; ri[t] = bi;
        __syncthreads();
        for (int off = 128; off > 0; off >>= 1) {
            if (t < off) {
                float ov = rv[t + off]; int oi = ri[t + off];
                if (ov < rv[t] || (ov == rv[t] && oi < ri[t])) { rv[t] = ov; ri[t] = oi; }
            }
            __syncthreads();
        }
        if (t == 0) {
            candVal[blockIdx.x * KSEL + iter] = rv[0];
            candIdx[blockIdx.x * KSEL + iter] = base + ri[0];
            vals[ri[0]] = __builtin_inff();
        }
        __syncthreads();
    }
}

// ---------------------------------------------------------------------------
// Kernel 4b: merge 2500 candidates -> final top-50 (values then indices).
// ---------------------------------------------------------------------------
__global__ void __launch_bounds__(256)
topk_phase_b(const float* __restrict__ candVal, const int* __restrict__ candIdx,
             float* __restrict__ out) {
    __shared__ float vals[NCAND];
    __shared__ int   gidx[NCAND];
    __shared__ float rv[256];
    __shared__ int   ri[256];
    __shared__ int   rp[256];
    const int t = threadIdx.x;
    for (int i = t; i < NCAND; i += 256) { vals[i] = candVal[i]; gidx[i] = candIdx[i]; }
    __syncthreads();

    for (int iter = 0; iter < KSEL; ++iter) {
        float bv = __builtin_inff();
        int   bg = 0x7fffffff;
        int   bp = 0;
        for (int i = t; i < NCAND; i += 256) {
            float vq = vals[i]; int g = gidx[i];
            if (vq < bv || (vq == bv && g < bg)) { bv = vq; bg = g; bp = i; }
        }
        rv[t] = bv; ri[t] = bg; rp[t] = bp;
        __syncthreads();
        for (int off = 128; off > 0; off >>= 1) {
            if (t < off) {
                if (rv[t + off] < rv[t] ||
                    (rv[t + off] == rv[t] && ri[t + off] < ri[t])) {
                    rv[t] = rv[t + off]; ri[t] = ri[t + off]; rp[t] = rp[t + off];
                }
            }
            __syncthreads();
        }
        if (t == 0) {
            out[iter]        = rv[0];
            out[KSEL + iter] = (float)ri[0];   // index slot of the flat float output
            vals[rp[0]]      = __builtin_inff();
        }
        __syncthreads();
    }
}

// ---------------------------------------------------------------------------
// Host-side launcher. Workspace layout (floats):
//   [0, 262144)          partial column sums (128 x 2048)        1 MB
//   [262144, 264192)     w                                        8 KB
//   [264192, 266240)     v                                        8 KB
//   [266240, 366240)     distances (100000)                     400 KB
//   [366240, 368740)     candidate values (2500)                 10 KB
//   [368740, 371240)     candidate indices (2500, int)           10 KB
// Total ~1.49 MB.
// ---------------------------------------------------------------------------
extern "C" void kernel_launch(void* const* d_in, const int* in_sizes, int n_in,
                              void* d_out, int out_size, void* d_ws, size_t ws_size,
                              hipStream_t stream) {
    const float* train = (const float*)d_in[0];
    const float* test  = (const float*)d_in[1];
    // d_in[2] is k (==50), folded into KSEL at compile time.

    float* ws      = (float*)d_ws;
    float* partial = ws;
    float* w       = partial + (size_t)RSLICES * DIM;
    float* v       = w + DIM;
    float* dist    = v + DIM;
    float* candVal = dist + N_ROWS;
    int*   candIdx = (int*)(candVal + NCAND);
    float* out     = (float*)d_out;

    colsumsq_partial<<<dim3(DIM / 256, RSLICES), 256, 0, stream>>>(train, partial);
    finalize_wv<<<1, 256, 0, stream>>>(partial, test, w, v);
    dist_wmma<<<(N_ROWS + 127) / 128, 256, 0, stream>>>(train, w, v, dist);
    topk_phase_a<<<NBLK_A, 256, 0, stream>>>(dist, candVal, candIdx);
    topk_phase_b<<<1, 256, 0, stream>>>(candVal, candIdx, out);
}